// StaticGrid_31353261261050
// MI455X (gfx1250) — compile-verified
//
#include <hip/hip_runtime.h>
#include <hip/hip_bf16.h>
#include <cstdint>
#include <cstddef>

typedef int v4i __attribute__((ext_vector_type(4)));

// ---------------------------------------------------------------------------
// StaticGrid.calc_gradient_vector_at_node
//   pass 0 (optional): pack node (x,y,z) into interleaved float3 (1 gather
//                      record instead of 3 per node reference)
//   pass 1 (per patch): cross-product normal -> (slope, x_slope, y_slope)
//   pass 2 (per node):  NaN-aware masked mean over <=8 patches
// Identities: cos(atan2(-uy,-ux)) = -nx/sqrt(nx^2+ny^2) (magnitude cancels),
// likewise sin; only acos() survives as a transcendental.
// ---------------------------------------------------------------------------

__device__ __forceinline__ void patch_math(
    float x0, float y0, float z0,
    float x1, float y1, float z1,
    float x2, float y2, float z2,
    float& slope, float& xs, float& ys)
{
    float pqx = x1 - x0, pqy = y1 - y0, pqz = z1 - z0;
    float prx = x2 - x0, pry = y2 - y0, prz = z2 - z0;
    // normal = pq x pr
    float nx = pqy * prz - pqz * pry;
    float ny = pqz * prx - pqx * prz;
    float nz = pqx * pry - pqy * prx;
    float m2 = nx * nx + ny * ny + nz * nz;
    float uz = nz * rsqrtf(m2);          // NaN for degenerate patches, like ref
    float s  = acosf(uz);                // slope_at_patch (NaN propagates)
    float h2 = nx * nx + ny * ny;
    float rh = (h2 > 0.0f) ? rsqrtf(h2) : 0.0f;
    slope = s;
    xs = -nx * rh * s;                   // cos(theta) * slope
    ys = -ny * rh * s;                   // sin(theta) * slope
}

// Stage this block's 3 KB slice of nodes_at_patch into LDS via the CDNA5
// async-to-LDS path (ASYNCcnt). Non-temporal: the index stream is read once
// and must not evict the L2-resident gather working sets.
__device__ __forceinline__ void stage_nap_async(
    const int* __restrict__ nap, int* s_idx, int t, int base)
{
    unsigned lds_off = (unsigned)(size_t)(&s_idx[0]) + (unsigned)t * 4u;
    unsigned g_off   = ((unsigned)(base * 3 + t)) * 4u;
    // The 24-bit instruction offset is added to BOTH global and LDS addresses,
    // so one (vdst, vaddr) pair covers the whole 3 KB tile in three issues.
    asm volatile(
        "global_load_async_to_lds_b32 %0, %1, %2 offset:0 th:TH_LOAD_NT\n\t"
        "global_load_async_to_lds_b32 %0, %1, %2 offset:1024 th:TH_LOAD_NT\n\t"
        "global_load_async_to_lds_b32 %0, %1, %2 offset:2048 th:TH_LOAD_NT\n\t"
        "s_wait_asynccnt 0\n\t"
        :
        : "v"(lds_off), "v"(g_off), "s"(nap)
        : "memory");
}

// ---------------------------------------------------------------------------
// Pass 0: pack node coords into interleaved float3 (kept RT -> L2 resident).
// Source streams are read-once -> non-temporal loads.
// ---------------------------------------------------------------------------
__global__ __launch_bounds__(256)
void pack_nodes_kernel(const float* __restrict__ node_x,
                       const float* __restrict__ node_y,
                       const float* __restrict__ z,
                       float*       __restrict__ packed,   // [n_nodes, 3] = x,y,z
                       int n_nodes)
{
    const int i = blockIdx.x * blockDim.x + threadIdx.x;
    if (i >= n_nodes) return;
    const float x = __builtin_nontemporal_load(node_x + i);
    const float y = __builtin_nontemporal_load(node_y + i);
    const float h = __builtin_nontemporal_load(z + i);
    float* rec = packed + (size_t)i * 3;
    rec[0] = x; rec[1] = y; rec[2] = h;   // RT store: keep resident for pass 1
}

// ---------------------------------------------------------------------------
// Pass 1 (packed tier): 3 x 12B gathers per patch from the L2-resident
// packed node buffer.
// ---------------------------------------------------------------------------
__global__ __launch_bounds__(256)
void patch_kernel_packed(const float* __restrict__ packed,    // [n_nodes, 3]
                         const int*   __restrict__ nap,       // [n_patches, 3]
                         float*       __restrict__ patch_out, // [n_patches, 3]
                         int n_patches)
{
    __shared__ int s_idx[256 * 3];
    const int t    = threadIdx.x;
    const int base = blockIdx.x * 256;
    const bool full = (base + 256 <= n_patches);   // uniform per block

    if (full) {
        stage_nap_async(nap, s_idx, t, base);
    } else {
        const int i = base + t;
        if (i < n_patches) {
            s_idx[t * 3 + 0] = nap[i * 3 + 0];
            s_idx[t * 3 + 1] = nap[i * 3 + 1];
            s_idx[t * 3 + 2] = nap[i * 3 + 2];
        }
    }
    __syncthreads();

    const int i = base + t;
    if (i >= n_patches) return;

    const float* r0 = packed + (size_t)s_idx[t * 3 + 0] * 3;
    const float* r1 = packed + (size_t)s_idx[t * 3 + 1] * 3;
    const float* r2 = packed + (size_t)s_idx[t * 3 + 2] * 3;

    float s, xs, ys;
    patch_math(r0[0], r0[1], r0[2],
               r1[0], r1[1], r1[2],
               r2[0], r2[1], r2[2],
               s, xs, ys);

    float* o = patch_out + (size_t)i * 3;   // RT store: resident for pass 2
    o[0] = s; o[1] = xs; o[2] = ys;
}

// ---------------------------------------------------------------------------
// Pass 1 (split tier): same but gathers from the three separate node arrays.
// ---------------------------------------------------------------------------
__global__ __launch_bounds__(256)
void patch_kernel_split(const float* __restrict__ z,
                        const float* __restrict__ node_x,
                        const float* __restrict__ node_y,
                        const int*   __restrict__ nap,
                        float*       __restrict__ patch_out,
                        int n_patches)
{
    __shared__ int s_idx[256 * 3];
    const int t    = threadIdx.x;
    const int base = blockIdx.x * 256;
    const bool full = (base + 256 <= n_patches);

    if (full) {
        stage_nap_async(nap, s_idx, t, base);
    } else {
        const int i = base + t;
        if (i < n_patches) {
            s_idx[t * 3 + 0] = nap[i * 3 + 0];
            s_idx[t * 3 + 1] = nap[i * 3 + 1];
            s_idx[t * 3 + 2] = nap[i * 3 + 2];
        }
    }
    __syncthreads();

    const int i = base + t;
    if (i >= n_patches) return;

    const int p0 = s_idx[t * 3 + 0];
    const int p1 = s_idx[t * 3 + 1];
    const int p2 = s_idx[t * 3 + 2];

    float s, xs, ys;
    patch_math(node_x[p0], node_y[p0], z[p0],
               node_x[p1], node_y[p1], z[p1],
               node_x[p2], node_y[p2], z[p2],
               s, xs, ys);

    float* o = patch_out + (size_t)i * 3;
    o[0] = s; o[1] = xs; o[2] = ys;
}

// ---------------------------------------------------------------------------
// Pass 2: patches_at_node streamed non-temporally (read-once, 128 MB) with
// global_prefetch ahead; patch records gathered (12B, L2-resident); output
// written non-temporally (write-once, never re-read).
// nanmean == sum/count with 0/0 -> NaN.
// ---------------------------------------------------------------------------
__global__ __launch_bounds__(256)
void node_kernel(const int*   __restrict__ pan,        // [n_nodes, 8]
                 const float* __restrict__ patch_out,  // [n_patches, 3]
                 float*       __restrict__ out,        // [n_nodes, 3]
                 int n_nodes)
{
    const int i = blockIdx.x * blockDim.x + threadIdx.x;
    if (i >= n_nodes) return;

    const v4i* pv = (const v4i*)pan + (size_t)i * 2;
    __builtin_prefetch(pv + 1024, 0, 0);   // stream ~16 KB ahead -> global_prefetch_b8

    const v4i a = __builtin_nontemporal_load(pv);
    const v4i b = __builtin_nontemporal_load(pv + 1);
    const int idx[8] = {a.x, a.y, a.z, a.w, b.x, b.y, b.z, b.w};

    float s0 = 0.f, s1 = 0.f, s2 = 0.f;
    float c0 = 0.f, c1 = 0.f, c2 = 0.f;
#pragma unroll
    for (int k = 0; k < 8; ++k) {
        const int p = idx[k];
        if (p >= 0) {
            const float* rec = patch_out + (size_t)p * 3;  // 12B gather
            const float s = rec[0];
            const float x = rec[1];
            const float y = rec[2];
            if (s == s) { s0 += s; c0 += 1.f; }
            if (x == x) { s1 += x; c1 += 1.f; }
            if (y == y) { s2 += y; c2 += 1.f; }
        }
    }
    float* o = out + (size_t)i * 3;
    __builtin_nontemporal_store(s0 / c0, o + 0);  // 0/0 -> NaN (empty nanmean)
    __builtin_nontemporal_store(s1 / c1, o + 1);
    __builtin_nontemporal_store(s2 / c2, o + 2);
}

// ---------------------------------------------------------------------------
// Zero-scratch fallback: fully fused per-node kernel.
// ---------------------------------------------------------------------------
__global__ __launch_bounds__(256)
void fused_kernel(const float* __restrict__ z,
                  const float* __restrict__ node_x,
                  const float* __restrict__ node_y,
                  const int*   __restrict__ nap,
                  const int*   __restrict__ pan,
                  float*       __restrict__ out,
                  int n_nodes)
{
    const int i = blockIdx.x * blockDim.x + threadIdx.x;
    if (i >= n_nodes) return;

    const v4i* pv = (const v4i*)pan + (size_t)i * 2;
    __builtin_prefetch(pv + 1024, 0, 0);
    const v4i a = __builtin_nontemporal_load(pv);
    const v4i b = __builtin_nontemporal_load(pv + 1);
    const int idx[8] = {a.x, a.y, a.z, a.w, b.x, b.y, b.z, b.w};

    float s0 = 0.f, s1 = 0.f, s2 = 0.f;
    float c0 = 0.f, c1 = 0.f, c2 = 0.f;
#pragma unroll
    for (int k = 0; k < 8; ++k) {
        const int p = idx[k];
        if (p >= 0) {
            const int p0 = nap[(size_t)p * 3 + 0];
            const int p1 = nap[(size_t)p * 3 + 1];
            const int p2 = nap[(size_t)p * 3 + 2];
            float s, xs, ys;
            patch_math(node_x[p0], node_y[p0], z[p0],
                       node_x[p1], node_y[p1], z[p1],
                       node_x[p2], node_y[p2], z[p2],
                       s, xs, ys);
            if (s  == s)  { s0 += s;  c0 += 1.f; }
            if (xs == xs) { s1 += xs; c1 += 1.f; }
            if (ys == ys) { s2 += ys; c2 += 1.f; }
        }
    }
    float* o = out + (size_t)i * 3;
    __builtin_nontemporal_store(s0 / c0, o + 0);
    __builtin_nontemporal_store(s1 / c1, o + 1);
    __builtin_nontemporal_store(s2 / c2, o + 2);
}

extern "C" void kernel_launch(void* const* d_in, const int* in_sizes, int n_in,
                              void* d_out, int out_size, void* d_ws, size_t ws_size,
                              hipStream_t stream)
{
    const float* z      = (const float*)d_in[0];
    const float* node_x = (const float*)d_in[1];
    const float* node_y = (const float*)d_in[2];
    const int*   nap    = (const int*)d_in[3];   // [n_patches, 3]
    const int*   pan    = (const int*)d_in[4];   // [n_nodes, 8]
    float*       out    = (float*)d_out;

    const int n_nodes   = in_sizes[0];
    const int n_patches = in_sizes[3] / 3;

    const size_t patch_bytes  = (size_t)n_patches * 3u * sizeof(float);
    const size_t packed_bytes = (size_t)n_nodes   * 3u * sizeof(float);

    const int pblk = (n_patches + 255) / 256;
    const int nblk = (n_nodes   + 255) / 256;

    if (ws_size >= patch_bytes + packed_bytes) {
        // Tier A: pack nodes -> packed-gather pass 1 -> pass 2
        float* pbuf   = (float*)d_ws;                       // [n_patches,3]
        float* packed = (float*)((char*)d_ws + patch_bytes);// [n_nodes,3]
        pack_nodes_kernel<<<nblk, 256, 0, stream>>>(node_x, node_y, z,
                                                    packed, n_nodes);
        patch_kernel_packed<<<pblk, 256, 0, stream>>>(packed, nap, pbuf,
                                                      n_patches);
        node_kernel<<<nblk, 256, 0, stream>>>(pan, pbuf, out, n_nodes);
    } else if (ws_size >= patch_bytes) {
        // Tier B: split-array pass 1 -> pass 2
        float* pbuf = (float*)d_ws;
        patch_kernel_split<<<pblk, 256, 0, stream>>>(z, node_x, node_y, nap,
                                                     pbuf, n_patches);
        node_kernel<<<nblk, 256, 0, stream>>>(pan, pbuf, out, n_nodes);
    } else {
        // Tier C: zero-scratch fused
        fused_kernel<<<nblk, 256, 0, stream>>>(z, node_x, node_y, nap, pan,
                                               out, n_nodes);
    }
}